// PGA_Unfold_J10_PC_AP_62036507623444
// MI455X (gfx1250) — compile-verified
//
#include <hip/hip_runtime.h>
#include <math.h>

// Problem constants (must match reference)
constexpr int KK  = 4;
constexpr int BB  = 16;
constexpr int NT  = 128;
constexpr int NRF = 16;
constexpr int MM  = 32;
constexpr int JJ  = 10;
#define LN2F 0.69314718055994530942f
#define EPSF 1e-12f

typedef __attribute__((ext_vector_type(2))) float v2f;
typedef __attribute__((ext_vector_type(8))) float v8f;

// D(16x16,f32) = A(16x4,f32) * B(4x16,f32) + C  -- CDNA5 V_WMMA_F32_16X16X4_F32
// A frag: lane L holds row M=L%16, k = (L/16)*2 + t   (t = vector element 0,1)
// B frag: lane L holds col N=L%16, k = (L/16)*2 + t
// C/D:    lane L element v holds row M = v + 8*(L/16), col N = L%16
__device__ __forceinline__ v8f wmma4(v2f a, v2f b, v8f c) {
  return __builtin_amdgcn_wmma_f32_16x16x4_f32(false, a, false, b, (short)0, c,
                                               false, false);
}

__global__ __launch_bounds__(256) void pga_unfold_kernel(
    const float* __restrict__ Hre_g, const float* __restrict__ Him_g,
    const float* __restrict__ Rre_g, const float* __restrict__ Rim_g,
    const float* __restrict__ Fre_g, const float* __restrict__ Fim_g,
    const float* __restrict__ Wre_g, const float* __restrict__ Wim_g,
    const float* __restrict__ ss_g, float* __restrict__ out) {
  // ---- LDS state (≈60 KB per workgroup; WGP has 320 KB) ----
  __shared__ float Hre[MM * NT], Him[MM * NT];       // H0[b]  (32x128)
  __shared__ float fre[KK][NT], fim[KK][NT];         // active f per K slice
  __shared__ float Wre[NRF * MM], Wim[NRF * MM];     // W0[b]  (16x32)
  __shared__ float Vre[NRF * NRF], Vim[NRF * NRF];   // radar covariance
  __shared__ float Cre[MM * MM], Cim[MM * MM];       // C (32x32)
  __shared__ float Ere[MM * NRF], Eim[MM * NRF];     // E = alpha*conj(D)
  __shared__ float alph[MM], tm1[MM];
  __shared__ float g2re[NT], g2im[NT];               // comm grad (matrix part)
  __shared__ float grre[NT], grim[NT];               // radar grad
  __shared__ float gfre[NT], gfim[NT];               // total comm grad
  __shared__ float red[256];
  __shared__ float sc_c2, sc_t1;

  const int tid  = threadIdx.x;
  const int b    = blockIdx.x;
  const int lane = tid & 31;
  const int w    = tid >> 5;  // wave id 0..7

  // ---------------- precompute phase ----------------
  for (int it = 0; it < 16; ++it) {
    int idx = tid + it * 256;  // m*128+i
    Hre[idx] = Hre_g[b * (MM * NT) + idx];
    Him[idx] = Him_g[b * (MM * NT) + idx];
  }
  {
    int idx = tid;
    Wre[idx] = Wre_g[b * (NRF * MM) + idx];
    Wim[idx] = Wim_g[b * (NRF * MM) + idx];
    idx += 256;
    Wre[idx] = Wre_g[b * (NRF * MM) + idx];
    Wim[idx] = Wim_g[b * (NRF * MM) + idx];
  }
  if (tid < NT) {
    int i = tid, r = i >> 3;
    for (int k = 0; k < KK; ++k) {
      size_t o  = (((size_t)(k * BB + b)) * NT + i) * NRF + r;
      fre[k][i] = Fre_g[o];
      fim[k][i] = Fim_g[o];
    }
  }
  {  // warm R[b] into L2/WGP$ (global_prefetch_b8); 256B line per thread
    const float* rb = Rre_g + (size_t)b * NT * NT;
    const float* ib = Rim_g + (size_t)b * NT * NT;
    __builtin_prefetch(rb + tid * 64, 0, 0);
    __builtin_prefetch(ib + tid * 64, 0, 0);
  }
  __syncthreads();
  {  // Frobenius norm of W0[b]
    int i0 = tid * 2;
    float p = Wre[i0] * Wre[i0] + Wim[i0] * Wim[i0];
    p += Wre[i0 + 1] * Wre[i0 + 1] + Wim[i0 + 1] * Wim[i0 + 1];
    red[tid] = p;
  }
  __syncthreads();
  if (tid == 0) {
    float s = 0.f;
    for (int t = 0; t < 256; ++t) s += red[t];
    float fro = sqrtf(s) + EPSF;
    sc_c2     = (1.0f / (float)NRF) / (fro * fro);  // PT = 1
  }
  __syncthreads();
  {  // V[r][s] = c2 * sum_m W0[r,m] conj(W0[s,m])
    int r = tid >> 4, s = tid & 15;
    float vr = 0.f, vi = 0.f;
    for (int m = 0; m < MM; ++m) {
      float ar = Wre[r * MM + m], ai = Wim[r * MM + m];
      float br = Wre[s * MM + m], bi = Wim[s * MM + m];
      vr += ar * br + ai * bi;
      vi += ai * br - ar * bi;
    }
    Vre[tid] = sc_c2 * vr;
    Vim[tid] = sc_c2 * vi;
  }
  __syncthreads();

  // ---------------- J unfolded PGA iterations ----------------
  for (int j = 0; j < JJ; ++j) {
    const float mu  = ss_g[2 * j];
    const float rho = ss_g[2 * j + 1];

    if (w < 4) {
      // C(32x32) = Heff(32x128) @ Wc(128x32), complex, f32 WMMA.
      // Heff[m,i] = H0[m,i]*conj(f_i) ; Wc[i,m'] = conj(W0[i/8,m'])
      const int mt = w >> 1, nt = w & 1;
      const int row  = mt * 16 + (lane & 15);
      const int colc = nt * 16 + (lane & 15);
      v8f cr = {}, ci = {};
      for (int ks = 0; ks < 32; ++ks) {
        const int i0 = ks * 4 + (lane >> 4) * 2;
        v2f ar, ai, nai, br, bi;
        for (int t = 0; t < 2; ++t) {
          int i    = i0 + t;
          float hr = Hre[row * NT + i], hi = Him[row * NT + i];
          float fr = fre[0][i], fi = fim[0][i];
          float re = hr * fr + hi * fi;
          float im = hi * fr - hr * fi;
          ar[t] = re; ai[t] = im; nai[t] = -im;
          int r = i >> 3;
          br[t] = Wre[r * MM + colc];
          bi[t] = -Wim[r * MM + colc];
        }
        cr = wmma4(ar, br, cr);
        cr = wmma4(nai, bi, cr);
        ci = wmma4(ar, bi, ci);
        ci = wmma4(ai, br, ci);
      }
      for (int v = 0; v < 8; ++v) {
        int rr = mt * 16 + v + 8 * (lane >> 4);
        Cre[rr * MM + colc] = cr[v];
        Cim[rr * MM + colc] = ci[v];
      }
    } else {
      // RAD(128x16) = R(128x128) @ FV(128x16); FV[j,r] = f_j * V[j/8,r]
      // grad_rad[i] = 2*RAD[i, i/8]
      const float* Rre_b = Rre_g + (size_t)b * NT * NT;
      const float* Rim_b = Rim_g + (size_t)b * NT * NT;
      const int colr = lane & 15;
      for (int mq = 0; mq < 2; ++mq) {
        const int mt  = (w - 4) * 2 + mq;
        const int row = mt * 16 + (lane & 15);
        v8f cr = {}, ci = {};
        for (int ks = 0; ks < 32; ++ks) {
          const int i0 = ks * 4 + (lane >> 4) * 2;
          v2f ar, ai, nai, br, bi;
          for (int t = 0; t < 2; ++t) {
            int jj    = i0 + t;
            ar[t]     = Rre_b[row * NT + jj];
            float aim = Rim_b[row * NT + jj];
            ai[t] = aim; nai[t] = -aim;
            float fr = fre[0][jj], fi = fim[0][jj];
            float vr = Vre[(jj >> 3) * NRF + colr];
            float vi = Vim[(jj >> 3) * NRF + colr];
            br[t] = fr * vr - fi * vi;
            bi[t] = fr * vi + fi * vr;
          }
          cr = wmma4(ar, br, cr);
          cr = wmma4(nai, bi, cr);
          ci = wmma4(ar, bi, ci);
          ci = wmma4(ai, br, ci);
        }
        // need col = i/8 = 2*mt + h, held by lane with lane%16 == 2*mt+h
        int h = lane >> 4;
        if ((lane & 15) == 2 * mt + h) {
          for (int v = 0; v < 8; ++v) {
            int i   = mt * 16 + v + 8 * h;
            grre[i] = 2.0f * cr[v];
            grim[i] = 2.0f * ci[v];
          }
        }
      }
    }
    __syncthreads();

    // scalars: q_mm, q_tl, term1, alpha
    if (tid < MM) {
      int m    = tid;
      float qt = 0.f;
      for (int mp = 0; mp < MM; ++mp) {
        float a = Cre[m * MM + mp], bb = Cim[m * MM + mp];
        qt += a * a + bb * bb;
      }
      float a  = Cre[m * MM + m], bb = Cim[m * MM + m];
      float qm = a * a + bb * bb;
      qt -= qm;
      float d1 = qm + qt + 1.0f;  // SIGMA2 = 1
      float d2 = qt + 1.0f;
      tm1[m]   = qm / (LN2F * d1);
      alph[m]  = 2.0f * qm / (LN2F * d1 * d2);
    }
    __syncthreads();

    // E[m,r] = alpha[m]*conj(D[m,r]); D = C@W0^T - diag correction
    for (int p = tid; p < MM * NRF; p += 256) {
      int m = p >> 4, r = p & 15;
      float dr = 0.f, di = 0.f;
      for (int mp = 0; mp < MM; ++mp) {
        float c0 = Cre[m * MM + mp], c1 = Cim[m * MM + mp];
        float w0 = Wre[r * MM + mp], w1 = Wim[r * MM + mp];
        dr += c0 * w0 - c1 * w1;
        di += c0 * w1 + c1 * w0;
      }
      {
        float c0 = Cre[m * MM + m], c1 = Cim[m * MM + m];
        float w0 = Wre[r * MM + m], w1 = Wim[r * MM + m];
        dr -= c0 * w0 - c1 * w1;
        di -= c0 * w1 + c1 * w0;
      }
      Ere[p] = alph[m] * dr;
      Eim[p] = -alph[m] * di;
    }
    if (tid == 0) {
      float s = 0.f;
      for (int m = 0; m < MM; ++m) s += tm1[m];
      sc_t1 = s;
    }
    __syncthreads();

    // G2(128x16) = H0^T(128x32) @ E(32x16); grad2[i] = G2[i, i/8]
    {
      const int rowi = w * 16 + (lane & 15);
      const int colr = lane & 15;
      v8f cr = {}, ci = {};
      for (int ks = 0; ks < 8; ++ks) {
        const int m0 = ks * 4 + (lane >> 4) * 2;
        v2f ar, ai, nai, br, bi;
        for (int t = 0; t < 2; ++t) {
          int m     = m0 + t;
          ar[t]     = Hre[m * NT + rowi];
          float aim = Him[m * NT + rowi];
          ai[t] = aim; nai[t] = -aim;
          br[t] = Ere[m * NRF + colr];
          bi[t] = Eim[m * NRF + colr];
        }
        cr = wmma4(ar, br, cr);
        cr = wmma4(nai, bi, cr);
        ci = wmma4(ar, bi, ci);
        ci = wmma4(ai, br, ci);
      }
      int h = lane >> 4;
      if ((lane & 15) == 2 * w + h) {
        for (int v = 0; v < 8; ++v) {
          int i   = w * 16 + v + 8 * h;
          g2re[i] = cr[v];
          g2im[i] = ci[v];
        }
      }
    }
    __syncthreads();

    if (tid < NT) {
      gfre[tid] = sc_t1 * fre[0][tid] - g2re[tid];
      gfim[tid] = sc_t1 * fim[0][tid] - g2im[tid];
    }
    __syncthreads();

    // update + unit-modulus projection, all K slices
    for (int p = tid; p < KK * NT; p += 256) {
      int k = p >> 7, i = p & 127;
      float gr  = fre[k][i] + mu * gfre[i] - rho * grre[i];
      float gi  = fim[k][i] + mu * gfim[i] - rho * grim[i];
      float mag = sqrtf(gr * gr + gi * gi);
      if (mag > EPSF) {
        fre[k][i] = gr / mag;
        fim[k][i] = gi / mag;
      } else {
        fre[k][i] = 0.f;
        fim[k][i] = 0.f;
      }
    }
    __syncthreads();
  }

  // ---------------- write output (2,K,B,NT,NRF) ----------------
  for (int pk = 0; pk < 8; ++pk) {
    int part = pk >> 2, k = pk & 3;
    size_t base = (((size_t)part * KK + k) * BB + b) * (size_t)(NT * NRF);
    for (int rep = 0; rep < 8; ++rep) {
      int lin = rep * 256 + tid;  // 0..2047
      int i = lin >> 4, r = lin & 15;
      float v = 0.0f;
      if (r == (i >> 3)) v = part ? fim[k][i] : fre[k][i];
      out[base + lin] = v;
    }
  }
}

extern "C" void kernel_launch(void* const* d_in, const int* in_sizes, int n_in,
                              void* d_out, int out_size, void* d_ws,
                              size_t ws_size, hipStream_t stream) {
  (void)in_sizes; (void)n_in; (void)out_size; (void)d_ws; (void)ws_size;
  pga_unfold_kernel<<<BB, 256, 0, stream>>>(
      (const float*)d_in[0], (const float*)d_in[1],  // H_re, H_im
      (const float*)d_in[2], (const float*)d_in[3],  // R_re, R_im
      (const float*)d_in[4], (const float*)d_in[5],  // F_re, F_im
      (const float*)d_in[6], (const float*)d_in[7],  // W_re, W_im
      (const float*)d_in[8],                         // step_size
      (float*)d_out);
}